// SAGENet_51908974739870
// MI455X (gfx1250) — compile-verified
//
#include <hip/hip_runtime.h>

// Problem constants (match reference)
#define N_NODES 50000
#define IN_C    128
#define HID_C   256
#define OUT_C   128
#define N_EDGES 800000

typedef __attribute__((ext_vector_type(2))) float v2f;
typedef __attribute__((ext_vector_type(8))) float v8f;

// ---------------------------------------------------------------------------
// Zero a float buffer (grid-stride). Workspace is NOT re-poisoned between
// timed replays, so accumulators must be zeroed at the start of every launch.
// ---------------------------------------------------------------------------
__global__ void zero_f32(float* __restrict__ p, long long n) {
  long long i = (long long)blockIdx.x * blockDim.x + threadIdx.x;
  long long stride = (long long)gridDim.x * blockDim.x;
  for (; i < n; i += stride) p[i] = 0.0f;
}

// ---------------------------------------------------------------------------
// Degree: deg[dst[e]] += 1 (fp32, non-returning atomic -> resolves in L2)
// ---------------------------------------------------------------------------
__global__ void degree_kernel(const long long* __restrict__ dst,
                              float* __restrict__ deg, int ne) {
  int i = blockIdx.x * blockDim.x + threadIdx.x;
  if (i < ne) atomicAdd(&deg[dst[i]], 1.0f);
}

// ---------------------------------------------------------------------------
// Scatter-add: agg[dst[e], :] += x[src[e], :]
// One wave32 per edge; 32 lanes x float4 cover 128 channels per pass.
// Loads are coalesced row reads; atomics are contiguous within the dst row.
// Whole working set is L2-resident (192 MB L2 >> 154 MB), so these atomics
// execute at L2 bandwidth, not HBM.
// ---------------------------------------------------------------------------
template <int C>
__global__ void scatter_add_kernel(const float* __restrict__ xin,
                                   const long long* __restrict__ src,
                                   const long long* __restrict__ dst,
                                   float* __restrict__ agg, int ne) {
  int wave = blockIdx.x * (blockDim.x >> 5) + (threadIdx.x >> 5);
  int lane = threadIdx.x & 31;
  if (wave >= ne) return;
  long long s = src[wave];
  long long d = dst[wave];
  const float* xi = xin + s * (long long)C;
  float* o = agg + d * (long long)C;
#pragma unroll
  for (int c0 = 0; c0 < C; c0 += 128) {
    int c = c0 + lane * 4;
    float4 v = *(const float4*)(xi + c);
    atomicAdd(o + c + 0, v.x);
    atomicAdd(o + c + 1, v.y);
    atomicAdd(o + c + 2, v.z);
    atomicAdd(o + c + 3, v.w);
  }
}

// ---------------------------------------------------------------------------
// Fused SAGE linear:  out = (agg * invdeg) @ Wl^T + x @ Wr^T + b  [+ ReLU]
// via V_WMMA_F32_16X16X4_F32 over the concatenated K dim (2*CIN), fp32-exact.
//
// One wave computes a 16(M) x 16(N) tile. 256-thread blocks = 8 waves, each
// wave owns a different N-tile (better A-row reuse in the WGP caches).
//
// ISA fragment layouts (CDNA5 ISA):
//   A 16x4 f32 : v0 lanes0-15 K=k0,  lanes16-31 K=k0+2 ; v1 -> K+1
//   B 4x16 f32 : v0 lanes0-15 K=k0,  lanes16-31 K=k0+2 ; v1 -> K+1 ; N=lane%16
//   C/D 16x16  : VGPR v, lane l -> M = v + 8*(l>=16), N = l%16
// ---------------------------------------------------------------------------
template <int CIN, int COUT, bool RELU>
__global__ __launch_bounds__(256)
void sage_gemm_wmma(const float* __restrict__ agg, const float* __restrict__ xin,
                    const float* __restrict__ Wl, const float* __restrict__ Wr,
                    const float* __restrict__ bias, const float* __restrict__ deg,
                    float* __restrict__ out) {
  const int lane = threadIdx.x & 31;
  const int wid  = threadIdx.x >> 5;
  const int r    = lane & 15;   // row-in-tile (A) / col-in-tile (B,D)
  const int kh   = lane >> 4;   // K half: 0 -> {k0,k0+1}, 1 -> {k0+2,k0+3}

  const int mtile = blockIdx.x;
  const int ntile = blockIdx.y * 8 + wid;

  const long long m = (long long)mtile * 16 + r;  // A row this lane streams
  const int       n = ntile * 16 + r;             // B column this lane streams

  const float* arow  = agg + m * CIN;
  const float* xrow  = xin + m * CIN;
  const float* wlrow = Wl + (long long)n * CIN;
  const float* wrrow = Wr + (long long)n * CIN;
  const float  invd  = 1.0f / fmaxf(deg[m], 1.0f);

  v8f acc = {0.f, 0.f, 0.f, 0.f, 0.f, 0.f, 0.f, 0.f};

  // First half of K: mean (= agg * invd) against Wl^T
#pragma unroll 4
  for (int k0 = 0; k0 < CIN; k0 += 4) {
    const int ka = k0 + kh * 2;
    v2f a, b;
    a.x = arow[ka] * invd;
    a.y = arow[ka + 1] * invd;
    b.x = wlrow[ka];
    b.y = wlrow[ka + 1];
    acc = __builtin_amdgcn_wmma_f32_16x16x4_f32(false, a, false, b,
                                                (short)0, acc, false, false);
  }
  // Second half of K: x against Wr^T
#pragma unroll 4
  for (int k0 = 0; k0 < CIN; k0 += 4) {
    const int ka = k0 + kh * 2;
    v2f a, b;
    a.x = xrow[ka];
    a.y = xrow[ka + 1];
    b.x = wrrow[ka];
    b.y = wrrow[ka + 1];
    acc = __builtin_amdgcn_wmma_f32_16x16x4_f32(false, a, false, b,
                                                (short)0, acc, false, false);
  }

  const float bv = bias[n];
#pragma unroll
  for (int v = 0; v < 8; ++v) {
    const long long mo = (long long)mtile * 16 + v + kh * 8;
    float val = acc[v] + bv;
    if (RELU) val = fmaxf(val, 0.0f);
    out[mo * COUT + n] = val;
  }
}

// ---------------------------------------------------------------------------
// Orchestration. Workspace layout (floats):
//   [0, 50000)                       deg
//   [50176, 50176+12.8M)             h   (N x HID_C)
//   [50176+12.8M, 50176+25.6M)       agg (reused: layer1 uses N*IN_C of it,
//                                         layer2 uses N*HID_C)
// Total ~102.6 MB.
// ---------------------------------------------------------------------------
extern "C" void kernel_launch(void* const* d_in, const int* in_sizes, int n_in,
                              void* d_out, int out_size, void* d_ws, size_t ws_size,
                              hipStream_t stream) {
  const float*     x   = (const float*)d_in[0];
  const long long* ei  = (const long long*)d_in[1];  // jnp.int64 edge_index [2,E]
  const float*     W1l = (const float*)d_in[2];
  const float*     b1  = (const float*)d_in[3];
  const float*     W1r = (const float*)d_in[4];
  const float*     W2l = (const float*)d_in[5];
  const float*     b2  = (const float*)d_in[6];
  const float*     W2r = (const float*)d_in[7];
  float*           out = (float*)d_out;

  const long long* src = ei;            // edge_index[0]
  const long long* dst = ei + N_EDGES;  // edge_index[1]

  float* ws   = (float*)d_ws;
  float* deg  = ws;
  float* h    = ws + 50176;  // aligned past deg
  float* aggb = h + (long long)N_NODES * HID_C;

  // ---- Layer 1 ----
  zero_f32<<<2048, 256, 0, stream>>>(deg, (long long)N_NODES);
  zero_f32<<<8192, 256, 0, stream>>>(aggb, (long long)N_NODES * IN_C);
  degree_kernel<<<(N_EDGES + 255) / 256, 256, 0, stream>>>(dst, deg, N_EDGES);
  scatter_add_kernel<IN_C><<<(N_EDGES + 7) / 8, 256, 0, stream>>>(x, src, dst,
                                                                  aggb, N_EDGES);
  {
    dim3 grid(N_NODES / 16, (HID_C / 16) / 8);  // 3125 x 2, 8 waves/block
    sage_gemm_wmma<IN_C, HID_C, true>
        <<<grid, 256, 0, stream>>>(aggb, x, W1l, W1r, b1, deg, h);
  }

  // ---- Layer 2 ----
  zero_f32<<<8192, 256, 0, stream>>>(aggb, (long long)N_NODES * HID_C);
  scatter_add_kernel<HID_C><<<(N_EDGES + 7) / 8, 256, 0, stream>>>(h, src, dst,
                                                                   aggb, N_EDGES);
  {
    dim3 grid(N_NODES / 16, (OUT_C / 16) / 8);  // 3125 x 1, 8 waves/block
    sage_gemm_wmma<HID_C, OUT_C, false>
        <<<grid, 256, 0, stream>>>(aggb, h, W2l, W2r, b2, deg, out);
  }
}